// VectorQuantizer_30391188586692
// MI455X (gfx1250) — compile-verified
//
#include <hip/hip_runtime.h>
#include <hip/hip_bf16.h>

// Problem constants (from reference)
#define DIM   256
#define TLEN  1024
#define NCODE 1024
#define NB    32
#define NROWS (NB * TLEN)              // 32768
#define NELEM (NROWS * DIM)            // 8388608
#define BETA  0.25f

typedef __attribute__((ext_vector_type(2))) float v2f;
typedef __attribute__((ext_vector_type(8))) float v8f;

// ---------------------------------------------------------------------------
// Kernel 1: codebook squared norms + zero the histogram (deterministic reset)
// ---------------------------------------------------------------------------
__global__ void vq_prep(const float* __restrict__ W,
                        float* __restrict__ wnorm,
                        int* __restrict__ hist) {
    int k = blockIdx.x * blockDim.x + threadIdx.x;
    if (k < NCODE) {
        const float* row = W + (size_t)k * DIM;
        float s = 0.f;
        #pragma unroll 8
        for (int d = 0; d < DIM; ++d) { float v = row[d]; s += v * v; }
        wnorm[k] = s;
        hist[k]  = 0;
    }
}

// ---------------------------------------------------------------------------
// Kernel 2: main fused kernel.
//  - block = 256 threads = 8 waves, owns 16 consecutive rows (same b, t0..t0+15)
//  - wave w owns codes [w*128, w*128+128) = 8 WMMA n-tiles of 16 codes
//  - fp32 WMMA 16x16x4 accumulates dot(x_row, w_code) exactly in fp32
//  - epilogue: argmin of wnorm[k] - 2*dot, cross-lane + cross-wave reduce,
//    gather codebook row, straight-through output, squared-error partial.
// ---------------------------------------------------------------------------
__launch_bounds__(256, 2)
__global__ void vq_main(const float* __restrict__ Z,      // (b, d, t) fp32
                        const float* __restrict__ W,      // (K, d) fp32
                        const float* __restrict__ wnorm,  // (K,)
                        int* __restrict__ hist,           // (K,)
                        float* __restrict__ out,          // (b, d, t) fp32
                        float* __restrict__ partials) {   // (gridDim.x,)
    __shared__ float red_v[16][8];
    __shared__ int   red_i[16][8];
    __shared__ int   row_code[16];
    __shared__ float red_sq[256];

    const int tid  = threadIdx.x;
    const int wave = tid >> 5;
    const int lane = tid & 31;
    const int lo   = lane & 15;
    const int hi   = lane >> 4;

    const int n0 = blockIdx.x * 16;       // 16 rows, all in one batch image
    const int b  = n0 >> 10;              // TLEN = 1024
    const int t0 = n0 & (TLEN - 1);

    const float* zb = Z + (size_t)b * DIM * TLEN;   // layout: [d][t], t contiguous
    const int kbase = wave * 128;

    // --- fp32 WMMA GEMM: dot products for 16 rows x 128 codes ---------------
    v8f acc[8];
    #pragma unroll
    for (int i = 0; i < 8; ++i) acc[i] = (v8f)0.0f;

    // A 16x4 fp32 layout: lanes 0-15 M=0..15 {v0:K=0, v1:K=1};
    //                     lanes 16-31 M=0..15 {v0:K=2, v1:K=3}
    const float* aptr = zb + (size_t)(2 * hi) * TLEN + t0 + lo;

    for (int d0 = 0; d0 < DIM; d0 += 4) {
        v2f A;
        A.x = aptr[(size_t)d0 * TLEN];
        A.y = aptr[(size_t)(d0 + 1) * TLEN];
        #pragma unroll
        for (int nt = 0; nt < 8; ++nt) {
            // B 4x16 fp32: lanes index N (code), halves/regs index K (d slice)
            const float* bp = W + (size_t)(kbase + nt * 16 + lo) * DIM + d0 + 2 * hi;
            v2f B = *(const v2f*)bp;   // contiguous pair -> global_load_b64
            acc[nt] = __builtin_amdgcn_wmma_f32_16x16x4_f32(
                false, A, false, B, (short)0, acc[nt], false, false);
        }
    }

    // --- per-lane argmin of  wnorm[k] - 2*dot  over this wave's 8 tiles -----
    // C/D layout: lanes 0-15 N=lane, reg r -> M=r; lanes 16-31 -> M=r+8
    float bestv[8];
    int   besti[8];
    #pragma unroll
    for (int r = 0; r < 8; ++r) { bestv[r] = 3.4e38f; besti[r] = 0; }

    #pragma unroll
    for (int nt = 0; nt < 8; ++nt) {
        const int kc = kbase + nt * 16 + lo;
        const float wn = wnorm[kc];
        #pragma unroll
        for (int r = 0; r < 8; ++r) {
            float s = wn - 2.0f * acc[nt][r];
            if (s < bestv[r]) { bestv[r] = s; besti[r] = kc; }   // nt asc -> first-idx ties
        }
    }

    // --- cross-lane reduce within each 16-lane half (wave32 shfl_xor) -------
    #pragma unroll
    for (int r = 0; r < 8; ++r) {
        float v = bestv[r];
        int   i = besti[r];
        #pragma unroll
        for (int off = 1; off < 16; off <<= 1) {
            float ov = __shfl_xor(v, off, 32);
            int   oi = __shfl_xor(i, off, 32);
            if (ov < v || (ov == v && oi < i)) { v = ov; i = oi; }
        }
        if (lo == 0) {
            int m = r + 8 * hi;           // row within tile
            red_v[m][wave] = v;
            red_i[m][wave] = i;
        }
    }
    __syncthreads();

    // --- cross-wave reduce: waves cover ascending code ranges ---------------
    if (tid < 16) {
        float v = red_v[tid][0];
        int   i = red_i[tid][0];
        #pragma unroll
        for (int w = 1; w < 8; ++w) {
            float ov = red_v[tid][w];
            int   oi = red_i[tid][w];
            if (ov < v || (ov == v && oi < i)) { v = ov; i = oi; }
        }
        row_code[tid] = i;
        atomicAdd(&hist[i], 1);           // integer atomic: deterministic
    }
    __syncthreads();

    // --- gather + straight-through output + squared-error partial -----------
    // thread -> row m = tid&15 (coalesced over t), d group = tid>>4 (16 d's)
    const int m  = tid & 15;
    const int dg = tid >> 4;
    const int code = row_code[m];
    const float* wrow = W + (size_t)code * DIM;

    float sq = 0.f;
    #pragma unroll
    for (int dd = 0; dd < 16; ++dd) {
        int d = dg * 16 + dd;
        size_t zoff = (size_t)b * DIM * TLEN + (size_t)d * TLEN + t0 + m;
        float zev  = Z[zoff];
        float wv   = wrow[d];
        float diff = wv - zev;            // z_q - z_e (used by both losses)
        out[zoff]  = zev + diff;          // ze + stopgrad(zq - ze), as reference computes
        sq += diff * diff;
    }
    red_sq[tid] = sq;
    __syncthreads();
    #pragma unroll
    for (int s = 128; s > 0; s >>= 1) {   // fixed-order tree: deterministic
        if (tid < s) red_sq[tid] += red_sq[tid + s];
        __syncthreads();
    }
    if (tid == 0) partials[blockIdx.x] = red_sq[0];
}

// ---------------------------------------------------------------------------
// Kernel 3: scalars — reg_loss = 1.25 * mse ; perplexity from histogram
// ---------------------------------------------------------------------------
__global__ void vq_final(const float* __restrict__ partials,  // 2048
                         const int* __restrict__ hist,        // 1024
                         float* __restrict__ out_scalars) {
    __shared__ float s_sq[1024];
    __shared__ float s_h[1024];
    int tid = threadIdx.x;

    s_sq[tid] = partials[tid] + partials[tid + 1024];

    float p = (float)hist[tid] * (1.0f / (float)NROWS);
    s_h[tid] = p * logf(p + 1e-10f);
    __syncthreads();

    for (int s = 512; s > 0; s >>= 1) {
        if (tid < s) { s_sq[tid] += s_sq[tid + s]; s_h[tid] += s_h[tid + s]; }
        __syncthreads();
    }
    if (tid == 0) {
        float mse = s_sq[0] / (float)NELEM;
        out_scalars[0] = (1.0f + BETA) * mse;   // q_loss + BETA*e_loss (same mse)
        out_scalars[1] = expf(-s_h[0]);         // perplexity
    }
}

// ---------------------------------------------------------------------------
extern "C" void kernel_launch(void* const* d_in, const int* in_sizes, int n_in,
                              void* d_out, int out_size, void* d_ws, size_t ws_size,
                              hipStream_t stream) {
    const float* Z = (const float*)d_in[0];   // z_e (32, 256, 1024)
    const float* W = (const float*)d_in[1];   // embed_weight (1024, 256)
    float* out = (float*)d_out;               // [z_q_out | reg_loss | perplexity]

    char* ws = (char*)d_ws;
    float* wnorm    = (float*)(ws);           // 4 KB
    int*   hist     = (int*)  (ws + 4096);    // 4 KB
    float* partials = (float*)(ws + 8192);    // 8 KB (2048 blocks)

    vq_prep <<<4, 256, 0, stream>>>(W, wnorm, hist);
    vq_main <<<NROWS / 16, 256, 0, stream>>>(Z, W, wnorm, hist, out, partials);
    vq_final<<<1, 1024, 0, stream>>>(partials, hist, out + NELEM);
}